// GNN_73624329388511
// MI455X (gfx1250) — compile-verified
//
#include <hip/hip_runtime.h>
#include <hip/hip_bf16.h>
#include <math.h>

#define DFEAT   128
#define NLAYERS 5
#define BN_EPS_   1e-5f
#define NORM_EPS_ 1e-12f

typedef float v2f __attribute__((ext_vector_type(2)));
typedef float v8f __attribute__((ext_vector_type(8)));

__device__ __forceinline__ float4 ld4(const float* p) { return *(const float4*)p; }
__device__ __forceinline__ void st4(float* p, float4 v) { *(float4*)p = v; }

// counts[n] = 1 (self loop)
__global__ void k_init_counts(float* counts, int n) {
    int i = blockIdx.x * blockDim.x + threadIdx.x;
    if (i < n) counts[i] = 1.0f;
}

// counts[dst[e]] += 1
__global__ void k_count_edges(const int* __restrict__ ei, float* counts, int E) {
    int e = blockIdx.x * blockDim.x + threadIdx.x;
    if (e < E) atomicAdd(&counts[ei[E + e]], 1.0f);
}

// h0 = x_emb1[x[:,0]] + x_emb2[x[:,1]]   (1 wave per node, float4 per lane)
__global__ void k_node_embed(const int* __restrict__ x,
                             const float* __restrict__ emb1,
                             const float* __restrict__ emb2,
                             float* __restrict__ h, int n) {
    int t = blockIdx.x * blockDim.x + threadIdx.x;
    int node = t >> 5;
    if (node >= n) return;
    int f = (t & 31) * 4;
    int a0 = x[node * 2 + 0];
    int a1 = x[node * 2 + 1];
    float4 v1 = ld4(emb1 + (size_t)a0 * DFEAT + f);
    float4 v2 = ld4(emb2 + (size_t)a1 * DFEAT + f);
    float4 o; o.x = v1.x + v2.x; o.y = v1.y + v2.y; o.z = v1.z + v2.z; o.w = v1.w + v2.w;
    st4(h + (size_t)node * DFEAT + f, o);
}

__global__ void k_zero_stats(float* gsum, float* gsumsq) {
    int t = threadIdx.x;
    if (t < DFEAT) gsum[t] = 0.0f;
    else if (t < 2 * DFEAT) gsumsq[t - DFEAT] = 0.0f;
}

// agg[n] = h[n] + edge_emb1[l][4] + edge_emb2[l][0]   (self-loop msg, also zero-init)
__global__ void k_init_agg(const float* __restrict__ h,
                           const float* __restrict__ e1l,
                           const float* __restrict__ e2l,
                           float* __restrict__ agg, int n) {
    int t = blockIdx.x * blockDim.x + threadIdx.x;
    int node = t >> 5;
    if (node >= n) return;
    int f = (t & 31) * 4;
    float4 hv = ld4(h + (size_t)node * DFEAT + f);
    float4 b  = ld4(e1l + 4 * DFEAT + f);   // bond type 4 (self loop)
    float4 d  = ld4(e2l + 0 * DFEAT + f);   // dir type 0
    float4 o; o.x = hv.x + b.x + d.x; o.y = hv.y + b.y + d.y;
              o.z = hv.z + b.z + d.z; o.w = hv.w + b.w + d.w;
    st4(agg + (size_t)node * DFEAT + f, o);
}

// one wave per edge: agg[dst] += h[src] + ee   (4 f32 atomics per lane)
__global__ void k_edge_scatter(const int* __restrict__ ei,
                               const int* __restrict__ ea,
                               const float* __restrict__ h,
                               const float* __restrict__ e1l,
                               const float* __restrict__ e2l,
                               float* __restrict__ agg, int E) {
    int w = (blockIdx.x * blockDim.x + threadIdx.x) >> 5;
    if (w >= E) return;
    int lane = threadIdx.x & 31;
    int src = ei[w];
    int dst = ei[E + w];
    int bt  = ea[w * 2 + 0];
    int dr  = ea[w * 2 + 1];
    int f = lane * 4;
    float4 m  = ld4(h   + (size_t)src * DFEAT + f);
    float4 eb = ld4(e1l + (size_t)bt  * DFEAT + f);
    float4 ed = ld4(e2l + (size_t)dr  * DFEAT + f);
    m.x += eb.x + ed.x; m.y += eb.y + ed.y; m.z += eb.z + ed.z; m.w += eb.w + ed.w;
    float* a = agg + (size_t)dst * DFEAT + f;
    atomicAdd(a + 0, m.x); atomicAdd(a + 1, m.y);
    atomicAdd(a + 2, m.z); atomicAdd(a + 3, m.w);
}

// agg = L2normalize(agg / counts)  in place; wave shuffle reduction for row norm
__global__ void k_row_norm(float* __restrict__ agg, const float* __restrict__ counts, int n) {
    int t = blockIdx.x * blockDim.x + threadIdx.x;
    int node = t >> 5;
    if (node >= n) return;
    int f = (t & 31) * 4;
    float inv = 1.0f / counts[node];
    float4 v = ld4(agg + (size_t)node * DFEAT + f);
    v.x *= inv; v.y *= inv; v.z *= inv; v.w *= inv;
    float s = v.x * v.x + v.y * v.y + v.z * v.z + v.w * v.w;
    #pragma unroll
    for (int off = 16; off > 0; off >>= 1) s += __shfl_xor(s, off, 32);
    float sc = 1.0f / fmaxf(sqrtf(s), NORM_EPS_);
    v.x *= sc; v.y *= sc; v.z *= sc; v.w *= sc;
    st4(agg + (size_t)node * DFEAT + f, v);
}

// Column sums / sum-of-squares over nodes via V_WMMA_F32_16X16X4_F32 with all-ones A.
// A = ones -> D rows are all the column sums of B, invariant to B's K-row slot order.
// One wave per 16-node tile; 8 column slices x 4 K-chunks x 2 wmma = 64 wmma/tile.
__global__ void k_col_stats(const float* __restrict__ y,
                            float* __restrict__ gsum, float* __restrict__ gsumsq,
                            int n) {
    int lane = threadIdx.x & 31;
    int tile = __builtin_amdgcn_readfirstlane((int)(blockIdx.x * 8 + (threadIdx.x >> 5)));
    int tiles = n >> 4;
    if (tile < tiles) {
        const float* base = y + (size_t)tile * 16 * DFEAT;
        v2f ones = {1.0f, 1.0f};
        int half = lane >> 4;          // 0: lanes 0-15, 1: lanes 16-31
        int col0 = lane & 15;
        #pragma unroll
        for (int s = 0; s < 8; ++s) {
            v8f acc = {}; v8f accsq = {};
            int col = s * 16 + col0;
            #pragma unroll
            for (int k = 0; k < 4; ++k) {
                // two distinct rows per lane-half per VGPR pair; exact K order irrelevant (A==1)
                int r0 = k * 4 + half * 2 + 0;
                int r1 = k * 4 + half * 2 + 1;
                v2f b;  b.x = base[r0 * DFEAT + col];  b.y = base[r1 * DFEAT + col];
                acc = __builtin_amdgcn_wmma_f32_16x16x4_f32(
                        false, ones, false, b, (short)0, acc, false, false);
                v2f b2; b2.x = b.x * b.x; b2.y = b.y * b.y;
                accsq = __builtin_amdgcn_wmma_f32_16x16x4_f32(
                        false, ones, false, b2, (short)0, accsq, false, false);
            }
            // D vgpr0, lanes 0-15: column sum for column s*16+lane (lanes 16-31 duplicate it)
            float cs = acc[0], cq = accsq[0];
            if (lane < 16) {
                atomicAdd(&gsum[s * 16 + lane], cs);
                atomicAdd(&gsumsq[s * 16 + lane], cq);
            }
        }
    } else if (tile == tiles) {
        int start = tiles << 4;
        if (start < n) {            // scalar tail for n % 16 rows (none when n%16==0)
            for (int r = start; r < n; ++r) {
                #pragma unroll
                for (int j = 0; j < 4; ++j) {
                    int c = lane + 32 * j;
                    float v = y[(size_t)r * DFEAT + c];
                    atomicAdd(&gsum[c], v);
                    atomicAdd(&gsumsq[c], v * v);
                }
            }
        }
    }
}

// h_out = (y - mu) * rsqrt(var + eps) * gamma + beta, optional ReLU
__global__ void k_bn_apply(const float* __restrict__ y,
                           const float* __restrict__ gsum, const float* __restrict__ gsumsq,
                           const float* __restrict__ gamma, const float* __restrict__ beta,
                           float* __restrict__ hout, int n, int do_relu) {
    int t = blockIdx.x * blockDim.x + threadIdx.x;
    int node = t >> 5;
    if (node >= n) return;
    int f = (t & 31) * 4;
    float invN = 1.0f / (float)n;
    float4 x  = ld4(y + (size_t)node * DFEAT + f);
    float4 s  = ld4(gsum + f);
    float4 q  = ld4(gsumsq + f);
    float4 g  = ld4(gamma + f);
    float4 bb = ld4(beta + f);
    float mu0 = s.x * invN, mu1 = s.y * invN, mu2 = s.z * invN, mu3 = s.w * invN;
    float i0 = rsqrtf(fmaxf(q.x * invN - mu0 * mu0, 0.0f) + BN_EPS_);
    float i1 = rsqrtf(fmaxf(q.y * invN - mu1 * mu1, 0.0f) + BN_EPS_);
    float i2 = rsqrtf(fmaxf(q.z * invN - mu2 * mu2, 0.0f) + BN_EPS_);
    float i3 = rsqrtf(fmaxf(q.w * invN - mu3 * mu3, 0.0f) + BN_EPS_);
    float4 o;
    o.x = (x.x - mu0) * i0 * g.x + bb.x;
    o.y = (x.y - mu1) * i1 * g.y + bb.y;
    o.z = (x.z - mu2) * i2 * g.z + bb.z;
    o.w = (x.w - mu3) * i3 * g.w + bb.w;
    if (do_relu) {
        o.x = fmaxf(o.x, 0.0f); o.y = fmaxf(o.y, 0.0f);
        o.z = fmaxf(o.z, 0.0f); o.w = fmaxf(o.w, 0.0f);
    }
    st4(hout + (size_t)node * DFEAT + f, o);
}

extern "C" void kernel_launch(void* const* d_in, const int* in_sizes, int n_in,
                              void* d_out, int out_size, void* d_ws, size_t ws_size,
                              hipStream_t stream) {
    const int*   x     = (const int*)  d_in[0];   // [N,2]
    const int*   ei    = (const int*)  d_in[1];   // [2,E]
    const int*   ea    = (const int*)  d_in[2];   // [E,2]
    const float* xe1   = (const float*)d_in[3];   // [35,128]
    const float* xe2   = (const float*)d_in[4];   // [3,128]
    const float* ee1   = (const float*)d_in[5];   // [L,6,128]
    const float* ee2   = (const float*)d_in[6];   // [L,3,128]
    const float* gamma = (const float*)d_in[7];   // [L,128]
    const float* beta  = (const float*)d_in[8];   // [L,128]

    int n = in_sizes[0] / 2;
    int E = in_sizes[1] / 2;

    float* out = (float*)d_out;
    size_t ND = (size_t)n * DFEAT;
    char* ws = (char*)d_ws;
    float* bufA   = (float*)ws; ws += ND * sizeof(float);
    float* agg    = (float*)ws; ws += ND * sizeof(float);
    float* counts = (float*)ws; ws += (size_t)n * sizeof(float);
    float* gsum   = (float*)ws; ws += DFEAT * sizeof(float);
    float* gsumsq = (float*)ws;

    int nodeThreads = n * 32;
    int nodeBlocks  = (nodeThreads + 255) / 256;
    int edgeBlocks  = (E + 7) / 8;          // 8 waves/block, 1 wave/edge

    k_init_counts<<<(n + 255) / 256, 256, 0, stream>>>(counts, n);
    k_count_edges<<<(E + 255) / 256, 256, 0, stream>>>(ei, counts, E);
    k_node_embed <<<nodeBlocks, 256, 0, stream>>>(x, xe1, xe2, bufA, n);

    int tiles = n >> 4;
    int wavesNeeded = tiles + ((n & 15) ? 1 : 0);
    int statBlocks = (wavesNeeded + 7) / 8;

    for (int l = 0; l < NLAYERS; ++l) {
        const float* hin  = (l & 1) ? out  : bufA;
        float*       hout = (l & 1) ? bufA : out;
        const float* e1l = ee1 + (size_t)l * 6 * DFEAT;
        const float* e2l = ee2 + (size_t)l * 3 * DFEAT;

        k_zero_stats  <<<1, 256, 0, stream>>>(gsum, gsumsq);
        k_init_agg    <<<nodeBlocks, 256, 0, stream>>>(hin, e1l, e2l, agg, n);
        k_edge_scatter<<<edgeBlocks, 256, 0, stream>>>(ei, ea, hin, e1l, e2l, agg, E);
        k_row_norm    <<<nodeBlocks, 256, 0, stream>>>(agg, counts, n);
        k_col_stats   <<<statBlocks, 256, 0, stream>>>(agg, gsum, gsumsq, n);
        k_bn_apply    <<<nodeBlocks, 256, 0, stream>>>(agg, gsum, gsumsq,
                                                       gamma + (size_t)l * DFEAT,
                                                       beta  + (size_t)l * DFEAT,
                                                       hout, n, (l < NLAYERS - 1) ? 1 : 0);
    }
}